// GraphPosteriorHead_55087250539271
// MI455X (gfx1250) — compile-verified
//
#include <hip/hip_runtime.h>
#include <hip/hip_bf16.h>

#define B_ 32
#define N_ 64
#define D_ 256
#define TWO_D 512
#define M_ROWS 2048   // B*N

typedef __attribute__((ext_vector_type(2))) float v2f;
typedef __attribute__((ext_vector_type(8))) float v8f;

__device__ __forceinline__ float gelu_f(float x) {
    // exact GELU: 0.5*x*(1+erf(x/sqrt(2)))
    return 0.5f * x * (1.0f + erff(x * 0.70710678118654752f));
}

__device__ __forceinline__ float sigmoid_f(float x) {
    return 1.0f / (1.0f + expf(-x));
}

__device__ __forceinline__ v8f wmma_f32(v2f a, v2f b, v8f c) {
    return __builtin_amdgcn_wmma_f32_16x16x4_f32(false, a, false, b, (short)0, c,
                                                 false, false);
}

// -------------------------------------------------------------------------
// Fused 2-layer MLP encoder: out = gelu(gelu([ve|ctx] @ w1 + b1) @ w2 + b2)
// One block = 16 rows of the 2048-row activation matrix. 16 waves, each wave
// owns one 16-wide N tile. A slab and hidden slab live in LDS; all matmul
// work is V_WMMA_F32_16X16X4_F32.
// -------------------------------------------------------------------------
__global__ __launch_bounds__(512) void encode_kernel(
    const float* __restrict__ ve, const float* __restrict__ ctx,
    const float* __restrict__ w1, const float* __restrict__ b1,
    const float* __restrict__ w2, const float* __restrict__ b2,
    float* __restrict__ out)
{
    __shared__ float A[16][TWO_D + 4];   // stride 516 -> conflict-free column reads
    __shared__ float H[16][D_ + 4];      // stride 260

    const int tile = blockIdx.x;         // 0..127
    const int tid  = threadIdx.x;

    // stage 0: build A = [ve_row | ctx_row] for 16 rows
    for (int idx = tid; idx < 16 * TWO_D; idx += 512) {
        int row = idx >> 9, col = idx & 511;
        int m = tile * 16 + row;                 // global row, b = m>>6
        float v = (col < D_) ? ve[m * D_ + col]
                             : ctx[(m >> 6) * D_ + (col - D_)];
        A[row][col] = v;
    }
    __syncthreads();

    const int wave = tid >> 5, lane = tid & 31;
    const int n0   = wave * 16;
    const int r    = lane & 15;
    const int koff = (lane >> 4) << 1;   // 0 or 2
    const int mhi  = (lane >> 4) << 3;   // 0 or 8

    // stage 1: h = gelu(A @ w1 + b1), K = 512
    v8f acc = {};
    for (int k = 0; k < TWO_D; k += 4) {
        v2f a  = { A[r][k + koff], A[r][k + koff + 1] };
        v2f bb = { w1[(k + koff) * D_ + n0 + r],
                   w1[(k + koff + 1) * D_ + n0 + r] };
        acc = wmma_f32(a, bb, acc);
    }
    {
        float bias = b1[n0 + r];
        #pragma unroll
        for (int v = 0; v < 8; ++v)
            H[v + mhi][n0 + r] = gelu_f(acc[v] + bias);
    }
    __syncthreads();

    // stage 2: out = gelu(H @ w2 + b2), K = 256
    v8f acc2 = {};
    for (int k = 0; k < D_; k += 4) {
        v2f a  = { H[r][k + koff], H[r][k + koff + 1] };
        v2f bb = { w2[(k + koff) * D_ + n0 + r],
                   w2[(k + koff + 1) * D_ + n0 + r] };
        acc2 = wmma_f32(a, bb, acc2);
    }
    {
        float bias = b2[n0 + r];
        #pragma unroll
        for (int v = 0; v < 8; ++v)
            out[(tile * 16 + v + mhi) * D_ + n0 + r] = gelu_f(acc2[v] + bias);
    }
}

// -------------------------------------------------------------------------
// Single linear layer (no bias/activation): out = [enc|ctx] @ w  (K = 512)
// Used for fi/fj projections of _predict_edges.
// -------------------------------------------------------------------------
__global__ __launch_bounds__(512) void lin512_kernel(
    const float* __restrict__ enc, const float* __restrict__ ctx,
    const float* __restrict__ w, float* __restrict__ out)
{
    __shared__ float A[16][TWO_D + 4];

    const int tile = blockIdx.x;
    const int tid  = threadIdx.x;

    for (int idx = tid; idx < 16 * TWO_D; idx += 512) {
        int row = idx >> 9, col = idx & 511;
        int m = tile * 16 + row;
        float v = (col < D_) ? enc[m * D_ + col]
                             : ctx[(m >> 6) * D_ + (col - D_)];
        A[row][col] = v;
    }
    __syncthreads();

    const int wave = tid >> 5, lane = tid & 31;
    const int n0   = wave * 16;
    const int r    = lane & 15;
    const int koff = (lane >> 4) << 1;
    const int mhi  = (lane >> 4) << 3;

    v8f acc = {};
    for (int k = 0; k < TWO_D; k += 4) {
        v2f a  = { A[r][k + koff], A[r][k + koff + 1] };
        v2f bb = { w[(k + koff) * D_ + n0 + r],
                   w[(k + koff + 1) * D_ + n0 + r] };
        acc = wmma_f32(a, bb, acc);
    }
    #pragma unroll
    for (int v = 0; v < 8; ++v)
        out[(tile * 16 + v + mhi) * D_ + n0 + r] = acc[v];
}

// -------------------------------------------------------------------------
// Pairwise edge logits: logits[b,i,j] = sum_d gelu(fi[b,i,d]+fj[b,j,d]+b1[d])*w2[d] + b2
// One block per (b,i); wave-per-j, lanes stride D, shuffle reduce.
// -------------------------------------------------------------------------
__global__ __launch_bounds__(256) void pairwise_kernel(
    const float* __restrict__ fi, const float* __restrict__ fj,
    const float* __restrict__ b1, const float* __restrict__ w2,
    const float* __restrict__ b2, float* __restrict__ logits)
{
    const int bi = blockIdx.x;            // 0..2047
    const int b  = bi >> 6, i = bi & 63;
    const int tid = threadIdx.x;

    __shared__ float t[D_];
    __shared__ float wl[D_];
    t[tid]  = fi[bi * D_ + tid] + b1[tid];
    wl[tid] = w2[tid];
    __syncthreads();

    const int wave = tid >> 5, lane = tid & 31;
    const float bias = b2[0];

    for (int j = wave; j < N_; j += 8) {
        const float* fjr = fj + (b * N_ + j) * D_;
        float acc = 0.f;
        #pragma unroll
        for (int q = 0; q < 8; ++q) {
            int d = lane + q * 32;
            acc += gelu_f(t[d] + fjr[d]) * wl[d];
        }
        #pragma unroll
        for (int off = 16; off >= 1; off >>= 1)
            acc += __shfl_xor(acc, off, 32);
        if (lane == 0) {
            float v = acc + bias;
            if (i == j) v = -1.0e9f;
            logits[(b * N_ + i) * N_ + j] = v;
        }
    }
}

// -------------------------------------------------------------------------
// Per-batch entropy of sigmoid(Wd) + tiny gate MLP -> gate[b]
// -------------------------------------------------------------------------
__global__ __launch_bounds__(256) void entropy_gate_kernel(
    const float* __restrict__ Wd, const float* __restrict__ gw1,
    const float* __restrict__ gb1, const float* __restrict__ gw2,
    const float* __restrict__ gb2, const int* __restrict__ n_samples,
    float* __restrict__ gate)
{
    const int b = blockIdx.x, tid = threadIdx.x;
    __shared__ float red[256];

    float s = 0.f;
    for (int idx = tid; idx < N_ * N_; idx += 256) {
        float p = sigmoid_f(Wd[b * N_ * N_ + idx]);
        const float e = 1e-8f;
        s += -(p * logf(p + e) + (1.f - p) * logf(1.f - p + e));
    }
    red[tid] = s;
    __syncthreads();
    for (int off = 128; off >= 1; off >>= 1) {
        if (tid < off) red[tid] += red[tid + off];
        __syncthreads();
    }
    if (tid == 0) {
        float ent = red[0] / (float)(N_ * N_);
        float ns  = fminf((float)n_samples[0] * 0.001f, 1.f);
        float cc  = 0.5f;
        float z   = gb2[0];
        for (int o = 0; o < 64; ++o) {
            float h = ns * gw1[o] + cc * gw1[64 + o] + ent * gw1[128 + o] + gb1[o];
            z += gelu_f(h) * gw2[o];
        }
        gate[b] = sigmoid_f(z);
    }
}

// -------------------------------------------------------------------------
// Blend logits, sigmoid -> W_posterior (d_out), and Asq = Wsig^2 for expm
// -------------------------------------------------------------------------
__global__ __launch_bounds__(256) void posterior_kernel(
    const float* __restrict__ Wd, const float* __restrict__ Wc,
    const float* __restrict__ gate, float* __restrict__ out,
    float* __restrict__ Asq)
{
    const int idx = blockIdx.x * 256 + threadIdx.x;     // < 131072
    const int b   = idx >> 12;
    float g   = gate[b];
    float wpl = g * Wd[idx] + (1.f - g) * Wc[idx];
    float wp  = sigmoid_f(wpl);
    out[idx]  = wp;
    Asq[idx]  = wp * wp;
}

// -------------------------------------------------------------------------
// trace(expm(A)) - N per batch via scaling-and-squaring, all WMMA in LDS.
// A entries in [0,1), |A|_1 <= 63 -> scale 2^-8 (|X| <= 0.25), 12-term
// Taylor (Horner), 8 squarings. 16 waves = 4x4 grid of 16x16 C tiles.
// -------------------------------------------------------------------------
__constant__ float c_invfact[13] = {
    1.0f, 1.0f, 0.5f, 1.6666666666666666e-1f, 4.1666666666666664e-2f,
    8.3333333333333332e-3f, 1.3888888888888889e-3f, 1.9841269841269841e-4f,
    2.4801587301587302e-5f, 2.7557319223985893e-6f, 2.7557319223985888e-7f,
    2.5052108385441720e-8f, 2.0876756987868100e-9f
};

__global__ __launch_bounds__(512) void expm_trace_kernel(
    const float* __restrict__ Asq, float* __restrict__ traces)
{
    __shared__ float X[64][68];   // stride 68 -> conflict-free column reads
    __shared__ float P[64][68];

    const int b = blockIdx.x, tid = threadIdx.x;

    for (int idx = tid; idx < N_ * N_; idx += 512) {
        int i = idx >> 6, j = idx & 63;
        X[i][j] = Asq[b * N_ * N_ + idx] * (1.0f / 256.0f);   // s = 8
        P[i][j] = (i == j) ? c_invfact[12] : 0.f;
    }
    __syncthreads();

    const int wave = tid >> 5, lane = tid & 31;
    const int m0   = (wave >> 2) * 16, n0 = (wave & 3) * 16;
    const int r    = lane & 15;
    const int koff = (lane >> 4) << 1;
    const int mhi  = (lane >> 4) << 3;

    // Horner: P <- X @ P + invfact[k-1]*I, k = 12..1
    for (int k = 12; k >= 1; --k) {
        v8f acc = {};
        for (int kk = 0; kk < 64; kk += 4) {
            v2f a  = { X[m0 + r][kk + koff], X[m0 + r][kk + koff + 1] };
            v2f bb = { P[kk + koff][n0 + r], P[kk + koff + 1][n0 + r] };
            acc = wmma_f32(a, bb, acc);
        }
        __syncthreads();
        float diag = c_invfact[k - 1];
        #pragma unroll
        for (int v = 0; v < 8; ++v) {
            int m = m0 + v + mhi, n = n0 + r;
            P[m][n] = acc[v] + ((m == n) ? diag : 0.f);
        }
        __syncthreads();
    }

    // squaring: P <- P @ P, 8 times
    for (int sq = 0; sq < 8; ++sq) {
        v8f acc = {};
        for (int kk = 0; kk < 64; kk += 4) {
            v2f a  = { P[m0 + r][kk + koff], P[m0 + r][kk + koff + 1] };
            v2f bb = { P[kk + koff][n0 + r], P[kk + koff + 1][n0 + r] };
            acc = wmma_f32(a, bb, acc);
        }
        __syncthreads();
        #pragma unroll
        for (int v = 0; v < 8; ++v)
            P[m0 + v + mhi][n0 + r] = acc[v];
        __syncthreads();
    }

    if (tid == 0) {
        float tr = 0.f;
        for (int i = 0; i < 64; ++i) tr += P[i][i];
        traces[b] = tr - (float)N_;
    }
}

__global__ void finalize_kernel(const float* __restrict__ traces,
                                float* __restrict__ out_acyc)
{
    if (threadIdx.x == 0) {
        float s = 0.f;
        for (int b = 0; b < B_; ++b) s += traces[b];
        out_acyc[0] = s / (float)B_;
    }
}

// -------------------------------------------------------------------------
extern "C" void kernel_launch(void* const* d_in, const int* in_sizes, int n_in,
                              void* d_out, int out_size, void* d_ws, size_t ws_size,
                              hipStream_t stream)
{
    const float* ds_emb  = (const float*)d_in[0];
    const float* ve      = (const float*)d_in[1];
    const float* cl_ctx  = (const float*)d_in[2];
    const float* dge_w1  = (const float*)d_in[3];
    const float* dge_b1  = (const float*)d_in[4];
    const float* dge_w2  = (const float*)d_in[5];
    const float* dge_b2  = (const float*)d_in[6];
    const float* cge_w1  = (const float*)d_in[7];
    const float* cge_b1  = (const float*)d_in[8];
    const float* cge_w2  = (const float*)d_in[9];
    const float* cge_b2  = (const float*)d_in[10];
    const float* dep_w1  = (const float*)d_in[11];
    const float* dep_b1  = (const float*)d_in[12];
    const float* dep_w2  = (const float*)d_in[13];
    const float* dep_b2  = (const float*)d_in[14];
    const float* cep_w1  = (const float*)d_in[15];
    const float* cep_b1  = (const float*)d_in[16];
    const float* cep_w2  = (const float*)d_in[17];
    const float* cep_b2  = (const float*)d_in[18];
    const float* gate_w1 = (const float*)d_in[19];
    const float* gate_b1 = (const float*)d_in[20];
    const float* gate_w2 = (const float*)d_in[21];
    const float* gate_b2 = (const float*)d_in[22];
    const int*   n_samp  = (const int*)d_in[23];

    float* ws = (float*)d_ws;
    const size_t ENC = (size_t)M_ROWS * D_;     // 524288
    const size_t MAT = (size_t)B_ * N_ * N_;    // 131072
    float* var_data  = ws;
    float* var_claim = var_data  + ENC;
    float* fi_d      = var_claim + ENC;
    float* fj_d      = fi_d      + ENC;
    float* fi_c      = fj_d      + ENC;
    float* fj_c      = fi_c      + ENC;
    float* Wd        = fj_c      + ENC;
    float* Wc        = Wd        + MAT;
    float* Asq       = Wc        + MAT;
    float* gate      = Asq       + MAT;
    float* traces    = gate      + B_;

    float* out_W    = (float*)d_out;            // (B,N,N) = 131072 floats
    float* out_acyc = out_W + MAT;              // scalar

    const int MT = M_ROWS / 16;                 // 128 row tiles

    // 1) encoders (fused 2-layer MLP, f32 WMMA)
    encode_kernel<<<MT, 512, 0, stream>>>(ve, ds_emb, dge_w1, dge_b1, dge_w2, dge_b2, var_data);
    encode_kernel<<<MT, 512, 0, stream>>>(ve, cl_ctx, cge_w1, cge_b1, cge_w2, cge_b2, var_claim);

    // 2) fi/fj projections (wi = w1[:2D], wj = w1[2D:])
    lin512_kernel<<<MT, 512, 0, stream>>>(var_data,  ds_emb, dep_w1,                 fi_d);
    lin512_kernel<<<MT, 512, 0, stream>>>(var_data,  ds_emb, dep_w1 + TWO_D * D_,    fj_d);
    lin512_kernel<<<MT, 512, 0, stream>>>(var_claim, cl_ctx, cep_w1,                 fi_c);
    lin512_kernel<<<MT, 512, 0, stream>>>(var_claim, cl_ctx, cep_w1 + TWO_D * D_,    fj_c);

    // 3) pairwise edge logits
    pairwise_kernel<<<M_ROWS, 256, 0, stream>>>(fi_d, fj_d, dep_b1, dep_w2, dep_b2, Wd);
    pairwise_kernel<<<M_ROWS, 256, 0, stream>>>(fi_c, fj_c, cep_b1, cep_w2, cep_b2, Wc);

    // 4) entropy + gate MLP
    entropy_gate_kernel<<<B_, 256, 0, stream>>>(Wd, gate_w1, gate_b1, gate_w2, gate_b2, n_samp, gate);

    // 5) blend + sigmoid -> W_posterior; Asq for expm
    posterior_kernel<<<(int)(MAT / 256), 256, 0, stream>>>(Wd, Wc, gate, out_W, Asq);

    // 6) batched trace(expm(A)) - N  (WMMA scaling-and-squaring)
    expm_trace_kernel<<<B_, 512, 0, stream>>>(Asq, traces);

    // 7) acyc = mean over batch
    finalize_kernel<<<1, 32, 0, stream>>>(traces, out_acyc);
}